// FoldSDF_55619826483275
// MI455X (gfx1250) — compile-verified
//
#include <hip/hip_runtime.h>

// Problem constants (match reference)
#define KNN   20
#define RES   64
#define NPTS  4096
#define NQ    8192
#define NB    4

typedef float v2f __attribute__((ext_vector_type(2)));
typedef float v8f __attribute__((ext_vector_type(8)));

__device__ __forceinline__ float3 load3(const float* __restrict__ p, int idx) {
    return make_float3(p[3 * idx + 0], p[3 * idx + 1], p[3 * idx + 2]);
}

// ---------------------------------------------------------------------------
// Kernel 1: kNN (top-20 of -squared-distance, self included) + EdgeConv-style
// graph feature: feat[:, :3] = max_k (nbr - x), feat[:, 3:] = x.
//
// The WMMA computes the affine form directly using the K=3 padding slot:
//   A[m][0..2] = 2*x_m ,  A[m][3] = 1
//   B[0..2][n] =   x_n ,  B[3][n] = -||x_n||^2
//   D[m][n]    = 2<x_m,x_n> - ||x_n||^2        (neg_dist + ||x_m||^2)
// so the per-candidate epilogue is a single subtract of the lane's own sq.
// One wave = 32 rows (two 16-row WMMA tiles). blockDim = 128 (4 waves).
// ---------------------------------------------------------------------------
__global__ __launch_bounds__(128) void knn_feat_kernel(
        const float* __restrict__ pts, float* __restrict__ feat) {
    const int lane    = threadIdx.x & 31;
    const int w       = threadIdx.x >> 5;          // wave in block (0..3)
    const int b       = blockIdx.y;                // batch
    const int rowBase = blockIdx.x * 128 + w * 32; // first of this wave's 32 rows
    const float* __restrict__ P = pts + (size_t)b * NPTS * 3;

    // inner-product staging tile, padded to stride 20 floats (16B-aligned rows)
    __shared__ __align__(16) float tile[4][32][20];

    const int  lm = lane & 15;
    const bool hi = lane >= 16;
    const int  rlo = hi ? 8 : 0;

    // --- A fragments (f32 16x4): lanes 0-15 hold K={0,1}, lanes 16-31 hold
    // K={2,3}, M = lane%16.  Coordinates pre-scaled by 2; K=3 slot = 1.
    float3 pa0 = load3(P, rowBase + lm);
    float3 pa1 = load3(P, rowBase + 16 + lm);
    v2f a0; a0[0] = hi ? 2.0f * pa0.z : 2.0f * pa0.x; a0[1] = hi ? 1.0f : 2.0f * pa0.y;
    v2f a1; a1[0] = hi ? 2.0f * pa1.z : 2.0f * pa1.x; a1[1] = hi ? 1.0f : 2.0f * pa1.y;

    // per-lane row point (this lane owns row rowBase+lane for top-k)
    float3 prow = load3(P, rowBase + lane);
    const float sqr = prow.x * prow.x + prow.y * prow.y + prow.z * prow.z;

    // register-resident top-20 (descending neg_dist), streamed bubble-insert
    float vals[KNN];
    int   idxs[KNN];
#pragma unroll
    for (int t = 0; t < KNN; ++t) { vals[t] = -1e30f; idxs[t] = 0; }

    float* __restrict__ myrow = &tile[w][lane][0];
    const v8f zc = {};

    float3 c = load3(P, lm);  // first column tile, software-pipelined
    for (int j0 = 0; j0 < NPTS; j0 += 16) {
        // prefetch next column tile (uniform branch; EXEC stays all-ones)
        float3 cn = c;
        if (j0 + 16 < NPTS) cn = load3(P, j0 + 16 + lm);

        // --- B fragment (f32 4x16): N = lane%16; lanes 0-15 hold K={0,1},
        // lanes 16-31 hold K={2,3}; K=3 slot carries -||x_n||^2.
        const float sqc = c.x * c.x + c.y * c.y + c.z * c.z;
        v2f bf; bf[0] = hi ? c.z : c.x; bf[1] = hi ? -sqc : c.y;

        v8f d0 = __builtin_amdgcn_wmma_f32_16x16x4_f32(
                     false, a0, false, bf, (short)0, zc, false, false);
        v8f d1 = __builtin_amdgcn_wmma_f32_16x16x4_f32(
                     false, a1, false, bf, (short)0, zc, false, false);

        // D layout: VGPR v holds (M = v + rlo, N = lane%16). Transpose via LDS
        // so each lane sees its own row. Same-wave DS ops are in-order (ISA),
        // so no barrier needed; pin compiler ordering only.
#pragma unroll
        for (int v = 0; v < 8; ++v) {
            tile[w][rlo + v][lm]      = d0[v];
            tile[w][16 + rlo + v][lm] = d1[v];
        }
        asm volatile("" ::: "memory");

        const float4* __restrict__ rowv = reinterpret_cast<const float4*>(myrow);
#pragma unroll
        for (int q = 0; q < 4; ++q) {
            float4 inn = rowv[q];
            float iv[4] = {inn.x, inn.y, inn.z, inn.w};
#pragma unroll
            for (int e = 0; e < 4; ++e) {
                const float nd = iv[e] - sqr;       // = neg squared distance
                const int   j  = j0 + q * 4 + e;
                // strict '>' + increasing-j streaming == top_k stable ties.
                // Branchless bubble-insert: candidate carries displaced entry
                // downward; lowers to a v_cmp + v_cndmask chain (no branches).
                if (nd > vals[KNN - 1]) {
                    float cv = nd;
                    int   ci = j;
#pragma unroll
                    for (int p = 0; p < KNN; ++p) {
                        const bool  sw = cv > vals[p];
                        const float tv = vals[p];
                        const int   ti = idxs[p];
                        vals[p] = sw ? cv : tv;
                        idxs[p] = sw ? ci : ti;
                        cv      = sw ? tv : cv;
                        ci      = sw ? ti : ci;
                    }
                }
            }
        }
        asm volatile("" ::: "memory");
        c = cn;
    }

    // graph feature: max over k of (nbr - x); channels 3..5 = x
    float m0 = -1e30f, m1 = -1e30f, m2 = -1e30f;
#pragma unroll
    for (int t = 0; t < KNN; ++t) {
        float3 nb = load3(P, idxs[t]);
        m0 = fmaxf(m0, nb.x - prow.x);
        m1 = fmaxf(m1, nb.y - prow.y);
        m2 = fmaxf(m2, nb.z - prow.z);
    }
    const int r = rowBase + lane;
    float* __restrict__ F = feat + ((size_t)b * NPTS + r) * 6;
    F[0] = m0; F[1] = m1; F[2] = m2;
    F[3] = prow.x; F[4] = prow.y; F[5] = prow.z;
}

// ---------------------------------------------------------------------------
// Trilinear corner helper (faithful to reference _corners)
// ---------------------------------------------------------------------------
struct Corners {
    int   ix[8], iy[8], iz[8];
    float w[8];
};

__device__ __forceinline__ void make_corners(float3 p, Corners& c) {
    const float fx = p.x * (float)RES, fy = p.y * (float)RES, fz = p.z * (float)RES;
    const int i0x = (int)floorf(fx), i0y = (int)floorf(fy), i0z = (int)floorf(fz);
    const int i1x = ((int)ceilf(fx)) & (RES - 1);
    const int i1y = ((int)ceilf(fy)) & (RES - 1);
    const int i1z = ((int)ceilf(fz)) & (RES - 1);
    const float inv = 1.0f / (float)RES;
    const float x0x = i0x * inv, x0y = i0y * inv, x0z = i0z * inv;
    const float x1x = (i0x + 1) * inv, x1y = (i0y + 1) * inv, x1z = (i0z + 1) * inv;
#pragma unroll
    for (int cc = 0; cc < 8; ++cc) {
        const int bx = (cc >> 2) & 1, by = (cc >> 1) & 1, bz = cc & 1;
        c.ix[cc] = bx ? i1x : i0x;
        c.iy[cc] = by ? i1y : i0y;
        c.iz[cc] = bz ? i1z : i0z;
        const float ox = bx ? x0x : x1x;
        const float oy = by ? x0y : x1y;
        const float oz = bz ? x0z : x1z;
        c.w[cc] = (fabsf(p.x - ox) * (float)RES) *
                  (fabsf(p.y - oy) * (float)RES) *
                  (fabsf(p.z - oz) * (float)RES);
    }
}

// ---------------------------------------------------------------------------
// Kernel 0: zero the 25 MB grid in workspace (harness poisons d_ws)
// ---------------------------------------------------------------------------
__global__ void zero_kernel(float4* __restrict__ g, int n4) {
    int i = blockIdx.x * blockDim.x + threadIdx.x;
    const int stride = gridDim.x * blockDim.x;
    for (; i < n4; i += stride) g[i] = make_float4(0.f, 0.f, 0.f, 0.f);
}

// ---------------------------------------------------------------------------
// Kernel 2: trilinear scatter-add of feat onto (B,R,R,R,6) grid (atomics in L2)
// ---------------------------------------------------------------------------
__global__ void rasterize_kernel(const float* __restrict__ pts,
                                 const float* __restrict__ feat,
                                 float* __restrict__ grid) {
    const int tid = blockIdx.x * blockDim.x + threadIdx.x;
    if (tid >= NB * NPTS) return;
    const int b = tid / NPTS;
    const int n = tid - b * NPTS;

    float3 p = load3(pts + (size_t)b * NPTS * 3, n);
    const float* __restrict__ F = feat + (size_t)tid * 6;
    float f[6];
#pragma unroll
    for (int q = 0; q < 6; ++q) f[q] = F[q];

    Corners c;
    make_corners(p, c);

    float* __restrict__ G = grid + (size_t)b * (RES * RES * RES * 6);
#pragma unroll
    for (int cc = 0; cc < 8; ++cc) {
        const int flat = ((c.ix[cc] * RES + c.iy[cc]) * RES + c.iz[cc]) * 6;
        const float wgt = c.w[cc];
#pragma unroll
        for (int q = 0; q < 6; ++q)
            atomicAdd(&G[flat + q], wgt * f[q]);
    }
}

// ---------------------------------------------------------------------------
// Kernel 3: trilinear gather at query points -> out (B, NQ, 6)
// ---------------------------------------------------------------------------
__global__ void interp_kernel(const float* __restrict__ query,
                              const float* __restrict__ grid,
                              float* __restrict__ out) {
    const int tid = blockIdx.x * blockDim.x + threadIdx.x;
    if (tid >= NB * NQ) return;
    const int b = tid / NQ;
    const int m = tid - b * NQ;

    float3 p = load3(query + (size_t)b * NQ * 3, m);
    Corners c;
    make_corners(p, c);

    const float* __restrict__ G = grid + (size_t)b * (RES * RES * RES * 6);
    float acc[6] = {0.f, 0.f, 0.f, 0.f, 0.f, 0.f};
#pragma unroll
    for (int cc = 0; cc < 8; ++cc) {
        const int   flat = ((c.ix[cc] * RES + c.iy[cc]) * RES + c.iz[cc]) * 6;
        const float wgt  = c.w[cc];
#pragma unroll
        for (int q = 0; q < 6; ++q)
            acc[q] += wgt * G[flat + q];
    }
    float* __restrict__ O = out + (size_t)tid * 6;
#pragma unroll
    for (int q = 0; q < 6; ++q) O[q] = acc[q];
}

// ---------------------------------------------------------------------------
extern "C" void kernel_launch(void* const* d_in, const int* in_sizes, int n_in,
                              void* d_out, int out_size, void* d_ws, size_t ws_size,
                              hipStream_t stream) {
    const float* pts   = (const float*)d_in[0]; // (4, 4096, 3) f32
    const float* query = (const float*)d_in[1]; // (4, 8192, 3) f32
    float*       out   = (float*)d_out;         // (4, 8192, 6) f32

    const size_t gridElems = (size_t)NB * RES * RES * RES * 6; // 6,291,456 f32
    float* grid = (float*)d_ws;
    float* feat = grid + gridElems;             // (4, 4096, 6) f32

    // 0) clear grid (workspace is poisoned)
    zero_kernel<<<1024, 256, 0, stream>>>((float4*)grid, (int)(gridElems / 4));

    // 1) kNN + graph feature (WMMA distance tiles)
    knn_feat_kernel<<<dim3(NPTS / 128, NB), 128, 0, stream>>>(pts, feat);

    // 2) trilinear scatter-add
    rasterize_kernel<<<(NB * NPTS + 255) / 256, 256, 0, stream>>>(pts, feat, grid);

    // 3) trilinear gather at queries
    interp_kernel<<<(NB * NQ + 255) / 256, 256, 0, stream>>>(query, grid, out);
}